// PNL_59356448031475
// MI455X (gfx1250) — compile-verified
//
#include <hip/hip_runtime.h>
#include <hip/hip_bf16.h>

// ---------------------------------------------------------------------------
// Non-local block (no softmax) rewritten via associativity:
//   y = (theta @ phi / N) @ g  ==  theta @ (phi @ g) / N
// folded end-to-end into per-batch 256x512 linear maps on x.
// GEMMs: V_WMMA_F32_16X16X4_F32 + double-buffered GLOBAL_LOAD_ASYNC_TO_LDS_B32.
// ---------------------------------------------------------------------------

#define NB    4
#define DIM   256
#define CHALF 128
#define HW    9216
#define NTOK  4608
#define BN_EPS 1e-5f

typedef __attribute__((ext_vector_type(2))) float v2f;
typedef __attribute__((ext_vector_type(8))) float v8f;

__device__ __forceinline__ v8f wmma_f32(v2f a, v2f b, v8f c) {
  // D = A(16x4) * B(4x16) + C, fp32
  return __builtin_amdgcn_wmma_f32_16x16x4_f32(false, a, false, b, (short)0, c,
                                               false, false);
}

// Raw 32-bit LDS byte offset of a __shared__ element (addrspace(3) cast).
typedef __attribute__((address_space(3))) float lds_f;
__device__ __forceinline__ unsigned ldsOff(float* p) {
  return (unsigned)(unsigned long long)(lds_f*)p;
}

// GVS-mode async copy: copies 4 bytes from global address base + goff to LDS byte offset ldsaddr; tracked by ASYNCcnt.
__device__ __forceinline__ void asyncLdsB32(unsigned ldsaddr, unsigned goff,
                                            const float* base) {
  asm volatile("global_load_async_to_lds_b32 %0, %1, %2"
               :
               : "v"(ldsaddr), "v"(goff), "s"(base)
               : "memory");
}
__device__ __forceinline__ void waitAsync12() {
  asm volatile("s_wait_asynccnt 12" ::: "memory");
}
__device__ __forceinline__ void waitAsync0() {
  asm volatile("s_wait_asynccnt 0" ::: "memory");
}

// ---------------------------------------------------------------------------
// K1: Out[b] (128 x 9216) = W (128x256) @ X[b] (256x9216) + bias
// block = 256 threads (8 waves). Tile 128x64, K chunk 16, double-buffered.
// ---------------------------------------------------------------------------
__global__ __launch_bounds__(256) void k_proj(const float* __restrict__ W,
                                              const float* __restrict__ bias,
                                              const float* __restrict__ X,
                                              float* __restrict__ Out) {
  const int b    = blockIdx.z;
  const int col0 = blockIdx.x * 64;
  const unsigned xOff = (unsigned)b * DIM * HW * 4u;
  float* Ob = Out + (size_t)b * CHALF * HW;

  __shared__ float As[2][128 * 16];
  __shared__ float Bs[2][64 * 16];

  const int tid = threadIdx.x;
  const int wave = tid >> 5, lane = tid & 31;
  const int lhalf = lane >> 4, lm = lane & 15;

  auto issue = [&](int buf, int kb) {
#pragma unroll
    for (int i = 0; i < 8; ++i) {
      int flat = tid + i * 256;
      int r = flat >> 4, k = flat & 15;
      asyncLdsB32(ldsOff(&As[buf][flat]), (unsigned)((r * DIM + kb + k) * 4),
                  W);
    }
#pragma unroll
    for (int i = 0; i < 4; ++i) {
      int flat = tid + i * 256;
      int n = flat & 63, k = flat >> 6;
      asyncLdsB32(ldsOff(&Bs[buf][n * 16 + k]),
                  (unsigned)(((kb + k) * HW + col0 + n) * 4) + xOff, X);
    }
  };

  v8f acc[4] = {};
  const int NCH = DIM / 16;  // 16 chunks
  issue(0, 0);
  for (int i = 0; i < NCH; ++i) {
    const int cur = i & 1;
    if (i + 1 < NCH) {
      issue(1 - cur, (i + 1) * 16);
      waitAsync12();
    } else {
      waitAsync0();
    }
    __syncthreads();
    const int arow = wave * 16 + lm;
#pragma unroll
    for (int kk = 0; kk < 16; kk += 4) {
      const int ko = kk + lhalf * 2;
      v2f a = *(const v2f*)&As[cur][arow * 16 + ko];
#pragma unroll
      for (int cg = 0; cg < 4; ++cg) {
        v2f bv = *(const v2f*)&Bs[cur][(cg * 16 + lm) * 16 + ko];
        acc[cg] = wmma_f32(a, bv, acc[cg]);
      }
    }
    __syncthreads();
  }

#pragma unroll
  for (int cg = 0; cg < 4; ++cg)
#pragma unroll
    for (int r = 0; r < 8; ++r) {
      int row = wave * 16 + lhalf * 8 + r;
      Ob[(size_t)row * HW + col0 + cg * 16 + lm] = acc[cg][r] + bias[row];
    }
}

// ---------------------------------------------------------------------------
// K2: M[b] (256x256) += PhXv[b](256x4608) @ GXv[b](256x4608)^T, K-split by 8.
// grid.z = b*8 + ksplit; epilogue atomicAdd into zeroed M.
// ---------------------------------------------------------------------------
__global__ __launch_bounds__(256) void k_gram(const float* __restrict__ Ph,
                                              const float* __restrict__ G,
                                              float* __restrict__ M) {
  const int b  = blockIdx.z >> 3;
  const int ks = blockIdx.z & 7;
  const int m0   = blockIdx.y * 128;
  const int col0 = blockIdx.x * 64;
  const unsigned bOff = (unsigned)b * DIM * NTOK * 4u;
  const int kstart = ks * 576;
  float* Mb = M + (size_t)b * DIM * DIM;

  __shared__ float As[2][128 * 16];
  __shared__ float Bs[2][64 * 16];

  const int tid = threadIdx.x;
  const int wave = tid >> 5, lane = tid & 31;
  const int lhalf = lane >> 4, lm = lane & 15;

  auto issue = [&](int buf, int kb) {
#pragma unroll
    for (int i = 0; i < 8; ++i) {
      int flat = tid + i * 256;
      int r = flat >> 4, k = flat & 15;
      asyncLdsB32(ldsOff(&As[buf][flat]),
                  (unsigned)(((m0 + r) * NTOK + kb + k) * 4) + bOff, Ph);
    }
#pragma unroll
    for (int i = 0; i < 4; ++i) {
      int flat = tid + i * 256;
      int n = flat >> 4, k = flat & 15;  // B^T[k,n]=GXv[col0+n,k]: k contiguous
      asyncLdsB32(ldsOff(&Bs[buf][flat]),
                  (unsigned)(((col0 + n) * NTOK + kb + k) * 4) + bOff, G);
    }
  };

  v8f acc[4] = {};
  const int NCH = 576 / 16;  // 36 chunks
  issue(0, kstart);
  for (int i = 0; i < NCH; ++i) {
    const int cur = i & 1;
    if (i + 1 < NCH) {
      issue(1 - cur, kstart + (i + 1) * 16);
      waitAsync12();
    } else {
      waitAsync0();
    }
    __syncthreads();
    const int arow = wave * 16 + lm;
#pragma unroll
    for (int kk = 0; kk < 16; kk += 4) {
      const int ko = kk + lhalf * 2;
      v2f a = *(const v2f*)&As[cur][arow * 16 + ko];
#pragma unroll
      for (int cg = 0; cg < 4; ++cg) {
        v2f bv = *(const v2f*)&Bs[cur][(cg * 16 + lm) * 16 + ko];
        acc[cg] = wmma_f32(a, bv, acc[cg]);
      }
    }
    __syncthreads();
  }

#pragma unroll
  for (int cg = 0; cg < 4; ++cg)
#pragma unroll
    for (int r = 0; r < 8; ++r) {
      int row = m0 + wave * 16 + lhalf * 8 + r;
      atomicAdd(&Mb[(size_t)row * DIM + col0 + cg * 16 + lm], acc[cg][r]);
    }
}

// ---------------------------------------------------------------------------
__global__ void k_zero(float* __restrict__ p, int n) {
  int i = blockIdx.x * blockDim.x + threadIdx.x;
  if (i < n) p[i] = 0.f;
}

// ---------------------------------------------------------------------------
// K3a: Me[b,d,o] = sum_c w_w[o,c] * M[b,d,2c] ; Mo with 2c+1
// ---------------------------------------------------------------------------
__global__ void k_fold_me(const float* __restrict__ M,
                          const float* __restrict__ w_w,
                          float* __restrict__ Me, float* __restrict__ Mo) {
  int idx = blockIdx.x * blockDim.x + threadIdx.x;  // NB*256*256
  if (idx >= NB * DIM * DIM) return;
  int b = idx >> 16, d = (idx >> 8) & 255, o = idx & 255;
  const float* Mb = M + ((size_t)b * DIM + d) * DIM;
  float se = 0.f, so = 0.f;
  for (int c = 0; c < CHALF; ++c) {
    float w = w_w[o * CHALF + c];
    se += w * Mb[2 * c];
    so += w * Mb[2 * c + 1];
  }
  Me[idx] = se;
  Mo[idx] = so;
}

// ---------------------------------------------------------------------------
// K3b: Wu[b,o,e2] = (1/N) * sum_c Me[b, 2c+par, o] * th_w[c, e]
//      (par = e2>>8, e = e2&255); Wv with Mo.
// ---------------------------------------------------------------------------
__global__ void k_fold_u(const float* __restrict__ Me,
                         const float* __restrict__ Mo,
                         const float* __restrict__ th_w,
                         float* __restrict__ Wu, float* __restrict__ Wv) {
  int idx = blockIdx.x * blockDim.x + threadIdx.x;  // NB*256*512
  if (idx >= NB * DIM * 512) return;
  int b = idx >> 17, o = (idx >> 9) & 255, e2 = idx & 511;
  int par = e2 >> 8, e = e2 & 255;
  const float s = 1.0f / (float)NTOK;
  float su = 0.f, sv = 0.f;
  for (int c = 0; c < CHALF; ++c) {
    float t = th_w[c * DIM + e];
    size_t mi = ((size_t)b * DIM + (2 * c + par)) * DIM + o;
    su += Me[mi] * t;
    sv += Mo[mi] * t;
  }
  Wu[idx] = su * s;
  Wv[idx] = sv * s;
}

// ---------------------------------------------------------------------------
// K3c: constU[b,o] = w_b[o] + (1/N) sum_c th_b[c]*(Me[b,2c,o]+Me[b,2c+1,o])
//      constV with Mo. Also zeroes the BN sum/sumsq accumulators.
// ---------------------------------------------------------------------------
__global__ void k_fold_c(const float* __restrict__ Me,
                         const float* __restrict__ Mo,
                         const float* __restrict__ th_b,
                         const float* __restrict__ w_b,
                         float* __restrict__ cU, float* __restrict__ cV,
                         float* __restrict__ stats) {
  int idx = blockIdx.x * blockDim.x + threadIdx.x;  // NB*256
  if (idx >= NB * DIM) return;
  int b = idx >> 8, o = idx & 255;
  const float s = 1.0f / (float)NTOK;
  float su = 0.f, sv = 0.f;
  for (int c = 0; c < CHALF; ++c) {
    float tb = th_b[c];
    size_t m0 = ((size_t)b * DIM + 2 * c) * DIM + o;
    size_t m1 = m0 + DIM;  // row 2c+1
    su += tb * (Me[m0] + Me[m1]);
    sv += tb * (Mo[m0] + Mo[m1]);
  }
  cU[idx] = w_b[o] + su * s;
  cV[idx] = w_b[o] + sv * s;
  if (idx < 512) stats[idx] = 0.f;  // zero sum[256] + sumsq[256]
}

// ---------------------------------------------------------------------------
// K4/K6: wy[b,o,p] = sum_k Wh[o,k] * xx[k,n] + const[o]
//   h = p/4608, n = p%4608, xx[k,n] = x[b, k&255, (k>>8)*4608 + n]
//   Wh = Wu (h=0) or Wv (h=1); Wv is contiguous after Wu so one SGPR base.
// FINAL=false: accumulate per-channel sum/sumsq (BN batch stats).
// FINAL=true : apply BN affine + residual, write d_out.
// ---------------------------------------------------------------------------
template <bool FINAL>
__global__ __launch_bounds__(256) void k_apply(
    const float* __restrict__ x, const float* __restrict__ Wu,
    const float* __restrict__ cU, const float* __restrict__ cV,
    float* __restrict__ stats, const float* __restrict__ bn_g,
    const float* __restrict__ bn_b, float* __restrict__ out) {
  const int b     = blockIdx.z;
  const int hhalf = blockIdx.y >> 1;
  const int m0    = (blockIdx.y & 1) * 128;
  const int col0  = blockIdx.x * 64;  // n within half

  const unsigned aOff =
      ((unsigned)hhalf * NB * DIM * 512u + (unsigned)b * DIM * 512u) * 4u;
  const unsigned xOff = (unsigned)b * DIM * HW * 4u;
  const float* cst = (hhalf ? cV : cU) + b * DIM;
  const float* xb  = x + (size_t)b * DIM * HW;

  __shared__ float As[2][128 * 16];
  __shared__ float Bs[2][64 * 16];

  const int tid = threadIdx.x;
  const int wave = tid >> 5, lane = tid & 31;
  const int lhalf = lane >> 4, lm = lane & 15;

  auto issue = [&](int buf, int kb) {
#pragma unroll
    for (int i = 0; i < 8; ++i) {
      int flat = tid + i * 256;
      int r = flat >> 4, k = flat & 15;
      asyncLdsB32(ldsOff(&As[buf][flat]),
                  (unsigned)(((m0 + r) * 512 + kb + k) * 4) + aOff, Wu);
    }
#pragma unroll
    for (int i = 0; i < 4; ++i) {
      int flat = tid + i * 256;
      int n = flat & 63, kg = kb + (flat >> 6);
      asyncLdsB32(
          ldsOff(&Bs[buf][n * 16 + (flat >> 6)]),
          (unsigned)(((kg & 255) * HW + (kg >> 8) * NTOK + col0 + n) * 4) +
              xOff,
          x);
    }
  };

  v8f acc[4] = {};
  const int NCH = 512 / 16;  // 32 chunks
  issue(0, 0);
  for (int i = 0; i < NCH; ++i) {
    const int cur = i & 1;
    if (i + 1 < NCH) {
      issue(1 - cur, (i + 1) * 16);
      waitAsync12();
    } else {
      waitAsync0();
    }
    __syncthreads();
    const int arow = wave * 16 + lm;
#pragma unroll
    for (int kk = 0; kk < 16; kk += 4) {
      const int ko = kk + lhalf * 2;
      v2f a = *(const v2f*)&As[cur][arow * 16 + ko];
#pragma unroll
      for (int cg = 0; cg < 4; ++cg) {
        v2f bv = *(const v2f*)&Bs[cur][(cg * 16 + lm) * 16 + ko];
        acc[cg] = wmma_f32(a, bv, acc[cg]);
      }
    }
    __syncthreads();
  }

  if (!FINAL) {
    float* gsum = stats;
    float* gss  = stats + 256;
#pragma unroll
    for (int r = 0; r < 8; ++r) {
      int ro = m0 + wave * 16 + lhalf * 8 + r;
      float c = cst[ro];
      float s1 = 0.f, s2 = 0.f;
#pragma unroll
      for (int cg = 0; cg < 4; ++cg) {
        float v = acc[cg][r] + c;
        s1 += v;
        s2 += v * v;
      }
      // reduce over the 16 lanes sharing this row (xor<=8 stays in half-wave)
      for (int m = 1; m < 16; m <<= 1) {
        s1 += __shfl_xor(s1, m, 32);
        s2 += __shfl_xor(s2, m, 32);
      }
      if (lm == 0) {
        atomicAdd(&gsum[ro], s1);
        atomicAdd(&gss[ro], s2);
      }
    }
  } else {
    const float* mean = stats + 512;
    const float* inv  = stats + 768;
    float* ob = out + (size_t)b * DIM * HW;
#pragma unroll
    for (int cg = 0; cg < 4; ++cg)
#pragma unroll
      for (int r = 0; r < 8; ++r) {
        int ro = m0 + wave * 16 + lhalf * 8 + r;
        int p = hhalf * NTOK + col0 + cg * 16 + lm;
        float wy = acc[cg][r] + cst[ro];
        ob[(size_t)ro * HW + p] =
            (wy - mean[ro]) * inv[ro] * bn_g[ro] + bn_b[ro] +
            xb[(size_t)ro * HW + p];
      }
  }
}

// ---------------------------------------------------------------------------
// K5: finalize batch stats -> mean, 1/sqrt(var+eps)
// ---------------------------------------------------------------------------
__global__ void k_stats_final(float* __restrict__ stats) {
  int o = threadIdx.x;  // 256
  const float cnt = (float)NB * (float)HW;
  float mean = stats[o] / cnt;
  float var = stats[256 + o] / cnt - mean * mean;
  stats[512 + o] = mean;
  stats[768 + o] = rsqrtf(var + BN_EPS);
}

// ---------------------------------------------------------------------------
extern "C" void kernel_launch(void* const* d_in, const int* in_sizes, int n_in,
                              void* d_out, int out_size, void* d_ws,
                              size_t ws_size, hipStream_t stream) {
  const float* x    = (const float*)d_in[0];
  const float* x_h  = (const float*)d_in[1];
  const float* g_w  = (const float*)d_in[2];
  const float* g_b  = (const float*)d_in[3];
  const float* th_w = (const float*)d_in[4];
  const float* th_b = (const float*)d_in[5];
  const float* ph_w = (const float*)d_in[6];
  const float* ph_b = (const float*)d_in[7];
  const float* w_w  = (const float*)d_in[8];
  const float* w_b  = (const float*)d_in[9];
  const float* bn_g = (const float*)d_in[10];
  const float* bn_b = (const float*)d_in[11];
  float* out = (float*)d_out;

  // workspace layout (floats); Wv MUST stay contiguous after Wu (k_apply).
  float* ws = (float*)d_ws;
  const size_t szProj = (size_t)NB * CHALF * HW;  // 4,718,592
  float* PhX = ws;
  float* GX  = PhX + szProj;
  float* M   = GX + szProj;                       // NB*256*256
  float* Me  = M + (size_t)NB * DIM * DIM;
  float* Mo  = Me + (size_t)NB * DIM * DIM;
  float* Wu  = Mo + (size_t)NB * DIM * DIM;       // NB*256*512
  float* Wv  = Wu + (size_t)NB * DIM * 512;       // contiguous after Wu
  float* cU  = Wv + (size_t)NB * DIM * 512;       // NB*256
  float* cV  = cU + (size_t)NB * DIM;
  float* stats = cV + (size_t)NB * DIM;           // 1024: sum|sumsq|mean|inv

  // 1-2) projections of x_h (WMMA GEMM, async double-buffered)
  dim3 gProj(HW / 64, 1, NB);
  k_proj<<<gProj, 256, 0, stream>>>(ph_w, ph_b, x_h, PhX);
  k_proj<<<gProj, 256, 0, stream>>>(g_w, g_b, x_h, GX);

  // 3) M = phi @ g  (256x256 per batch, K=4608 split by 8, atomic accumulate)
  k_zero<<<(NB * DIM * DIM) / 256, 256, 0, stream>>>(M, NB * DIM * DIM);
  dim3 gGram(DIM / 64, DIM / 128, NB * 8);
  k_gram<<<gGram, 256, 0, stream>>>(PhX, GX, M);

  // 4-6) fold M with w_w / th_w / biases into per-batch 256x512 maps
  k_fold_me<<<(NB * DIM * DIM) / 256, 256, 0, stream>>>(M, w_w, Me, Mo);
  k_fold_u<<<(NB * DIM * 512) / 256, 256, 0, stream>>>(Me, Mo, th_w, Wu, Wv);
  k_fold_c<<<(NB * DIM) / 256, 256, 0, stream>>>(Me, Mo, th_b, w_b, cU, cV,
                                                 stats);

  // 7) GEMM pass 1: BN batch statistics (sum / sumsq per channel)
  dim3 gApply(NTOK / 64, 4, NB);
  k_apply<false><<<gApply, 256, 0, stream>>>(x, Wu, cU, cV, stats, bn_g, bn_b,
                                             out);

  // 8) finalize mean / inv-std
  k_stats_final<<<1, 256, 0, stream>>>(stats);

  // 9) GEMM pass 2: BN affine + residual -> d_out
  k_apply<true><<<gApply, 256, 0, stream>>>(x, Wu, cU, cV, stats, bn_g, bn_b,
                                            out);
}